// AttentionOperation_15298673509077
// MI455X (gfx1250) — compile-verified
//
#include <hip/hip_runtime.h>
#include <math.h>
#include <stdint.h>

// Shapes from the reference
#define NB 4
#define HH 8
#define DD 64          // head dim (D == C == 64)
#define LL 2048

typedef __attribute__((ext_vector_type(2))) float v2f;
typedef __attribute__((ext_vector_type(8))) float v8f;

#define LDS_STRIDE 68   // 64 + 4 pad: bank-conflict-free, keeps 8B/16B alignment

__device__ __forceinline__ v8f wmma_f32(v2f a, v2f b, v8f c) {
    // D = A(16x4) * B(4x16) + C(16x16), pure fp32 (exact vs fp32 reference)
    return __builtin_amdgcn_wmma_f32_16x16x4_f32(false, a, false, b, (short)0, c, false, false);
}

// Generic pointer (to a __shared__ object) -> raw LDS byte offset for async ops.
__device__ __forceinline__ unsigned lds_off(const void* p) {
    return (unsigned)(uintptr_t)(const __attribute__((address_space(3))) void*)p;
}

// CDNA5 async DMA: 16 bytes global -> LDS per lane, tracked by ASYNCcnt.
__device__ __forceinline__ void async_copy16(unsigned ldsByteOff, const void* gaddr) {
    asm volatile("global_load_async_to_lds_b128 %0, %1, off"
                 :: "v"(ldsByteOff), "v"((uint64_t)(uintptr_t)gaddr)
                 : "memory");
}

// ---------------------------------------------------------------------------
// Kernel A: per-(b,h) stats of sim = Q^T K without materializing sim.
//   sum(sim)  = sum_d qsum[d]*ksum[d]
//   sum(sim^2)= <Q Q^T, K K^T>_F   (two 64x64 Gram matrices via WMMA)
// ---------------------------------------------------------------------------
__global__ __launch_bounds__(256) void stats1_kernel(const float* __restrict__ Q,
                                                     const float* __restrict__ K,
                                                     float* __restrict__ sum_bh,
                                                     float* __restrict__ ss_bh) {
    __shared__ float G[2 * 64 * 64];   // Gq | Gk
    __shared__ float rowp[2][64][2];   // row-sum partials
    __shared__ float qk[64];
    __shared__ float red[256];

    const int bh   = blockIdx.x;
    const int tid  = threadIdx.x;
    const int w    = tid >> 5;
    const int lane = tid & 31;
    const int lo   = lane & 15;
    const int hi   = lane >> 4;

    const size_t base = (size_t)bh * DD * LL;
    const float* Qb = Q + base;
    const float* Kb = K + base;

    for (int t = 0; t < 4; ++t) {
        const int tile = w * 4 + t;        // 0..31
        const int mat  = tile >> 4;        // 0 -> Gq, 1 -> Gk
        const int ti   = (tile >> 2) & 3;
        const int tj   = tile & 3;
        const float* src = mat ? Kb : Qb;
        const float* pa = src + (size_t)(16 * ti + lo) * LL;
        const float* pb = src + (size_t)(16 * tj + lo) * LL;
        v8f c8 = {};
        #pragma unroll 8
        for (int kk = 0; kk < 512; ++kk) {
            const int lcol = 4 * kk + 2 * hi;
            v2f a = *(const v2f*)(pa + lcol);   // A[M=d][K=l]
            v2f b = *(const v2f*)(pb + lcol);   // B[K=l][N=d']
            c8 = wmma_f32(a, b, c8);
        }
        #pragma unroll
        for (int r = 0; r < 8; ++r)
            G[mat * 4096 + (16 * ti + r + 8 * hi) * 64 + 16 * tj + lo] = c8[r];
    }

    {   // row sums
        const int mat  = (tid >= 128) ? 1 : 0;
        const int t2   = tid & 127;
        const int d    = t2 & 63;
        const int half = t2 >> 6;
        const float* src = mat ? Kb : Qb;
        const float* p = src + (size_t)d * LL + half * 1024;
        float acc = 0.f;
        for (int l = 0; l < 1024; ++l) acc += p[l];
        rowp[mat][d][half] = acc;
    }
    __syncthreads();
    if (tid < 64) {
        float qs = rowp[0][tid][0] + rowp[0][tid][1];
        float ks = rowp[1][tid][0] + rowp[1][tid][1];
        qk[tid] = qs * ks;
    }
    __syncthreads();

    float part = 0.f;
    #pragma unroll
    for (int j = 0; j < 16; ++j) {
        const int idx = tid * 16 + j;
        part += G[idx] * G[4096 + idx];
    }
    red[tid] = part;
    __syncthreads();
    for (int s = 128; s > 0; s >>= 1) {
        if (tid < s) red[tid] += red[tid + s];
        __syncthreads();
    }
    if (tid == 0) {
        float qksum = 0.f;
        for (int j = 0; j < 64; ++j) qksum += qk[j];
        sum_bh[bh] = qksum;
        ss_bh[bh]  = red[0];
    }
}

// ---------------------------------------------------------------------------
// Kernel B: combine per-h partials -> per-batch softmax scale s = rsqrt(var+eps)
// ---------------------------------------------------------------------------
__global__ void reduce1_kernel(const float* __restrict__ sum_bh,
                               const float* __restrict__ ss_bh,
                               float* __restrict__ sOut) {
    const int b = threadIdx.x;
    if (b < NB) {
        double s = 0.0, q = 0.0;
        for (int h = 0; h < HH; ++h) { s += sum_bh[b * HH + h]; q += ss_bh[b * HH + h]; }
        const double cnt  = (double)HH * (double)LL * (double)LL;
        const double mean = s / cnt;
        const double var  = q / cnt - mean * mean;
        sOut[b] = (float)(1.0 / sqrt(var + 1e-5));
    }
}

// ---------------------------------------------------------------------------
// Kernel C: fused flash attention, BL=64 queries / WG, BM=64 keys / iteration.
// K and V blocks are double-buffered in LDS via GLOBAL_LOAD_ASYNC_TO_LDS_B128
// (issue block i+1, s_wait_asynccnt 8 -> block i landed, barrier, compute).
// Each wave owns 2 S/E tiles (ti, tj0/tj1) and 2 O tiles (oc, ol0/ol1).
// No softmax rescaling needed (|s*sim| <~ 6 for N(0,1) inputs).
// ---------------------------------------------------------------------------
__global__ __launch_bounds__(256) void flash_kernel(const float* __restrict__ Q,
                                                    const float* __restrict__ K,
                                                    const float* __restrict__ V,
                                                    const float* __restrict__ sScale,
                                                    float* __restrict__ out,
                                                    float* __restrict__ psum,
                                                    float* __restrict__ pss) {
    __shared__ __attribute__((aligned(16))) float Kbuf[2][64 * LDS_STRIDE];
    __shared__ __attribute__((aligned(16))) float Vbuf[2][64 * LDS_STRIDE];
    __shared__ __attribute__((aligned(16))) float Elds[64 * LDS_STRIDE];
    __shared__ float denomLds[64];
    __shared__ float red[256];

    const int bh   = blockIdx.y;           // 0..31
    const int b    = bh >> 3;
    const int l0   = blockIdx.x * 64;
    const int tid  = threadIdx.x;
    const int w    = tid >> 5;
    const int lane = tid & 31;
    const int lo   = lane & 15;
    const int hi   = lane >> 4;
    const int ti  = w & 3;                 // S tile row (l)
    const int tj0 = (w >> 2) * 2, tj1 = tj0 + 1;   // S tile cols (m)
    const int oc  = w & 3;                 // O tile row (c)
    const int ol0 = (w >> 2) * 2, ol1 = ol0 + 1;   // O tile cols (l)
    const float s = sScale[b];

    const size_t base = (size_t)bh * DD * LL;
    const float* Qb = Q + base;
    const float* Kb = K + base;
    const float* Vb = V + base;

    // async-copy assignment: round j covers rows j*16..j*16+15, 4 floats/thread
    const int cpRow = tid >> 4;            // 0..15
    const int cpCol = (tid & 15) * 4;      // 0..60
    const unsigned kBase[2] = { lds_off(&Kbuf[0][0]), lds_off(&Kbuf[1][0]) };
    const unsigned vBase[2] = { lds_off(&Vbuf[0][0]), lds_off(&Vbuf[1][0]) };

    // Q fragments for this wave's S tiles (shared by tj0/tj1), kept in VGPRs.
    v2f qa[16];
    {
        const int lq = l0 + 16 * ti + lo;
        #pragma unroll
        for (int kk = 0; kk < 16; ++kk) {
            const int d = 4 * kk + 2 * hi;
            v2f a;
            a.x = Qb[(size_t)d * LL + lq];
            a.y = Qb[(size_t)(d + 1) * LL + lq];
            qa[kk] = a;
        }
    }

    v8f o8a = {}, o8b = {};
    float dA[8], dB[8];
    #pragma unroll
    for (int r = 0; r < 8; ++r) { dA[r] = 0.f; dB[r] = 0.f; }

    // prologue: issue async batch 0
    #pragma unroll
    for (int j = 0; j < 4; ++j) {
        const int d = j * 16 + cpRow;
        const unsigned loff = (unsigned)((d * LDS_STRIDE + cpCol) * 4);
        async_copy16(kBase[0] + loff, Kb + (size_t)d * LL + cpCol);
        async_copy16(vBase[0] + loff, Vb + (size_t)d * LL + cpCol);
    }

    for (int it = 0; it < LL / 64; ++it) {
        const int m0  = it * 64;
        const int par = it & 1;

        if (it + 1 < LL / 64) {            // issue next block, wait for current
            const int nxt = (it + 1) & 1;
            #pragma unroll
            for (int j = 0; j < 4; ++j) {
                const int d = j * 16 + cpRow;
                const unsigned loff = (unsigned)((d * LDS_STRIDE + cpCol) * 4);
                async_copy16(kBase[nxt] + loff, Kb + (size_t)d * LL + m0 + 64 + cpCol);
                async_copy16(vBase[nxt] + loff, Vb + (size_t)d * LL + m0 + 64 + cpCol);
            }
            asm volatile("s_wait_asynccnt 8" ::: "memory");
        } else {
            asm volatile("s_wait_asynccnt 0" ::: "memory");
        }
        __syncthreads();

        // ---- phase 1: two S tiles from LDS K, E = exp(s*S) ----
        const float* kb = &Kbuf[par][0];
        v8f c8a = {}, c8b = {};
        const int mS0 = 16 * tj0 + lo;
        const int mS1 = 16 * tj1 + lo;
        #pragma unroll
        for (int kk = 0; kk < 16; ++kk) {
            const int d = 4 * kk + 2 * hi;
            v2f k0, k1;
            k0.x = kb[d * LDS_STRIDE + mS0];
            k0.y = kb[(d + 1) * LDS_STRIDE + mS0];
            k1.x = kb[d * LDS_STRIDE + mS1];
            k1.y = kb[(d + 1) * LDS_STRIDE + mS1];
            c8a = wmma_f32(qa[kk], k0, c8a);
            c8b = wmma_f32(qa[kk], k1, c8b);
        }
        #pragma unroll
        for (int r = 0; r < 8; ++r) {
            const int row = (16 * ti + r + 8 * hi) * LDS_STRIDE;
            const float e0 = __expf(s * c8a[r]);
            const float e1 = __expf(s * c8b[r]);
            dA[r] += e0;
            dB[r] += e1;
            Elds[row + 16 * tj0 + lo] = e0;
            Elds[row + 16 * tj1 + lo] = e1;
        }
        __syncthreads();

        // ---- phase 2: two O tiles, O += V_blk * E^T (all operands in LDS) ----
        const float* vb = &Vbuf[par][0];
        #pragma unroll
        for (int kk = 0; kk < 16; ++kk) {
            const int m = 4 * kk + 2 * hi;
            v2f va = *(const v2f*)(vb + (16 * oc + lo) * LDS_STRIDE + m);      // V[c][m]
            v2f e0 = *(const v2f*)(&Elds[(16 * ol0 + lo) * LDS_STRIDE + m]);   // E[l][m]
            v2f e1 = *(const v2f*)(&Elds[(16 * ol1 + lo) * LDS_STRIDE + m]);
            o8a = wmma_f32(va, e0, o8a);
            o8b = wmma_f32(va, e1, o8b);
        }
        __syncthreads();
    }

    // ---- deterministic denominator reduction: 64 partials per query row ----
    #pragma unroll
    for (int r = 0; r < 8; ++r) {
        const int row = (16 * ti + r + 8 * hi) * LDS_STRIDE;
        Elds[row + 16 * tj0 + lo] = dA[r];
        Elds[row + 16 * tj1 + lo] = dB[r];
    }
    __syncthreads();
    if (tid < 64) {
        float acc = 0.f;
        for (int j = 0; j < 64; ++j) acc += Elds[tid * LDS_STRIDE + j];
        denomLds[tid] = 1.0f / acc;
    }
    __syncthreads();

    // ---- normalize, store rv (into d_out), accumulate LN2 partials ----
    const float inv0 = denomLds[16 * ol0 + lo];
    const float inv1 = denomLds[16 * ol1 + lo];
    const int lout0 = l0 + 16 * ol0 + lo;
    const int lout1 = l0 + 16 * ol1 + lo;
    float ps = 0.f, p2 = 0.f;
    #pragma unroll
    for (int r = 0; r < 8; ++r) {
        const size_t crow = base + (size_t)(16 * oc + r + 8 * hi) * LL;
        const float v0 = o8a[r] * inv0;
        const float v1 = o8b[r] * inv1;
        out[crow + lout0] = v0;
        out[crow + lout1] = v1;
        ps += v0 + v1;
        p2 += v0 * v0 + v1 * v1;
    }
    red[tid] = ps;
    __syncthreads();
    for (int st = 128; st > 0; st >>= 1) { if (tid < st) red[tid] += red[tid + st]; __syncthreads(); }
    const float blockSum = red[0];
    __syncthreads();
    red[tid] = p2;
    __syncthreads();
    for (int st = 128; st > 0; st >>= 1) { if (tid < st) red[tid] += red[tid + st]; __syncthreads(); }
    if (tid == 0) {
        psum[bh * 32 + blockIdx.x] = blockSum;
        pss[bh * 32 + blockIdx.x]  = red[0];
    }
}

// ---------------------------------------------------------------------------
// Kernel D: per-batch LN2 stats from 256 block partials
// ---------------------------------------------------------------------------
__global__ __launch_bounds__(256) void reduce2_kernel(const float* __restrict__ psum,
                                                      const float* __restrict__ pss,
                                                      float* __restrict__ m2,
                                                      float* __restrict__ i2) {
    __shared__ float r1[256], r2[256];
    const int b = blockIdx.x, tid = threadIdx.x;
    r1[tid] = psum[b * 256 + tid];
    r2[tid] = pss[b * 256 + tid];
    __syncthreads();
    for (int st = 128; st > 0; st >>= 1) {
        if (tid < st) { r1[tid] += r1[tid + st]; r2[tid] += r2[tid + st]; }
        __syncthreads();
    }
    if (tid == 0) {
        const double cnt  = (double)(HH * DD) * (double)LL;   // 512*2048
        const double mean = (double)r1[0] / cnt;
        const double var  = (double)r2[0] / cnt - mean * mean;
        m2[b] = (float)mean;
        i2[b] = (float)(1.0 / sqrt(var + 1e-5));
    }
}

// ---------------------------------------------------------------------------
// Kernel E: in-place LayerNorm2 + exact GELU over d_out
// ---------------------------------------------------------------------------
__global__ __launch_bounds__(256) void finalize_kernel(float* __restrict__ out,
                                                       const float* __restrict__ m2,
                                                       const float* __restrict__ i2) {
    const int idx = blockIdx.x * 256 + threadIdx.x;   // 4*512*2048 total
    const int b = idx >> 20;                          // 512*2048 = 2^20
    const float x = (out[idx] - m2[b]) * i2[b];
    out[idx] = 0.5f * x * (1.0f + erff(x * 0.70710678f));
}

extern "C" void kernel_launch(void* const* d_in, const int* in_sizes, int n_in,
                              void* d_out, int out_size, void* d_ws, size_t ws_size,
                              hipStream_t stream) {
    const float* Q = (const float*)d_in[0];
    const float* K = (const float*)d_in[1];
    const float* V = (const float*)d_in[2];
    float* out = (float*)d_out;
    float* ws  = (float*)d_ws;

    float* sum_bh = ws;               // 32
    float* ss_bh  = ws + 32;          // 32
    float* sScale = ws + 64;          // 4
    float* psum   = ws + 128;         // 1024
    float* pss    = ws + 128 + 1024;  // 1024
    float* m2     = ws + 128 + 2048;  // 4
    float* i2     = m2 + 4;           // 4   (~9 KB of workspace)

    stats1_kernel<<<dim3(NB * HH), 256, 0, stream>>>(Q, K, sum_bh, ss_bh);
    reduce1_kernel<<<1, 32, 0, stream>>>(sum_bh, ss_bh, sScale);
    flash_kernel<<<dim3(LL / 64, NB * HH), 256, 0, stream>>>(Q, K, V, sScale, out, psum, pss);
    reduce2_kernel<<<NB, 256, 0, stream>>>(psum, pss, m2, i2);
    finalize_kernel<<<(NB * HH * DD * LL) / 256, 256, 0, stream>>>(out, m2, i2);
}